// Upsample_74818330296431
// MI455X (gfx1250) — compile-verified
//
#include <hip/hip_runtime.h>

typedef __attribute__((ext_vector_type(2))) float v2f;
typedef __attribute__((ext_vector_type(8))) float v8f;

#define BB 8
#define NN 16384
#define MM 4096
#define CS 256
#define CF 128
#define TQ 128          // queries per block = 8 waves * 16 queries/wave
#define THREADS 256
#define NTILE (MM / 16) // 256 WMMA tiles over the sample set

// Branchless sorted-top-3 insert (d0 <= d1 <= d2), exact f32 compares.
__device__ __forceinline__ void insert3(float d, int i,
    float& d0, float& d1, float& d2, int& i0, int& i1, int& i2) {
  const bool lt2 = d < d2;
  const bool lt1 = d < d1;
  const bool lt0 = d < d0;
  d2 = lt1 ? d1 : (lt2 ? d : d2);
  i2 = lt1 ? i1 : (lt2 ? i : i2);
  d1 = lt0 ? d0 : (lt1 ? d : d1);
  i1 = lt0 ? i0 : (lt1 ? i : i1);
  d0 = lt0 ? d  : d0;
  i0 = lt0 ? i  : i0;
}

// Phase 1: 3-NN via V_WMMA_F32_16X16X4_F32.
// A (16x4, per sample row): (sx, sy, sz, |s|^2)
// B (4x16, per query col):  (-2qx, -2qy, -2qz, 1)
// A*B = -2 q.s + |s|^2 ; distance = A*B + |q|^2 (per-lane constant).
__global__ __launch_bounds__(THREADS)
void knn_wmma_kernel(const float* __restrict__ xyz,
                     const float* __restrict__ sxyz,
                     int* __restrict__ nidx,
                     float* __restrict__ nw) {
  extern __shared__ char smem[];
  float*  s_raw = (float*)smem;                     // 48 KB raw (x,y,z)*M
  float4* s_smp = (float4*)(smem + MM * 3 * 4);     // 64 KB (x,y,z,|s|^2)

  const int b    = blockIdx.x / (NN / TQ);
  const int qblk = blockIdx.x % (NN / TQ);
  const int tid  = threadIdx.x;

  // --- Stage the whole sample block HBM -> LDS with async-to-LDS DMA ---
  const float* sb = sxyz + (size_t)b * MM * 3;      // 16B aligned (49152*b)
  for (int i = tid; i < (MM * 3) / 4; i += THREADS) {
    unsigned     ldsoff = (unsigned)(i * 16);       // dyn-LDS base == offset 0
    const float* gaddr  = sb + i * 4;
    asm volatile("global_load_async_to_lds_b128 %0, %1, off"
                 :: "v"(ldsoff), "v"(gaddr) : "memory");
  }
  asm volatile("s_wait_asynccnt 0x0" ::: "memory");
  __syncthreads();

  // Build (x,y,z,|s|^2) table from the raw LDS copy.
  for (int i = tid; i < MM; i += THREADS) {
    float x = s_raw[i * 3 + 0];
    float y = s_raw[i * 3 + 1];
    float z = s_raw[i * 3 + 2];
    s_smp[i] = make_float4(x, y, z, x * x + y * y + z * z);
  }
  __syncthreads();

  const int  lane = tid & 31;
  const int  wv   = tid >> 5;
  const int  ln16 = lane & 15;
  const bool hi   = lane >= 16;      // lanes 16-31 carry K=2,3 / rows M=8..15

  const int qn = qblk * TQ + wv * 16 + ln16;
  const float* qp = xyz + ((size_t)b * NN + qn) * 3;
  const float qx = qp[0], qy = qp[1], qz = qp[2];
  const float q2 = qx * qx + qy * qy + qz * qz;

  v2f bmat;                          // B-matrix, column = this lane's query
  bmat.x = hi ? (-2.0f * qz) : (-2.0f * qx);   // K=2 : K=0
  bmat.y = hi ? 1.0f         : (-2.0f * qy);   // K=3 : K=1

  // Per-lane A operand pointer: half-select folded into the base, b64 loads.
  const v2f* ap = (const v2f*)s_smp + (ln16 * 2 + (hi ? 1 : 0));

  float d0 = 1e30f, d1 = 1e30f, d2 = 1e30f;
  int   i0 = 0,     i1 = 0,     i2 = 0;
  const int rbase = hi ? 8 : 0;      // C/D rows owned by this lane half

  v2f a_cur = ap[0];
  for (int t = 0; t < NTILE; ++t) {
    // Software-pipelined prefetch of next tile's A fragment.
    const int tn = (t + 1 < NTILE) ? (t + 1) : t;
    v2f a_nxt = ap[tn * 32];

    v8f c = {};
    v8f acc = __builtin_amdgcn_wmma_f32_16x16x4_f32(
        false, a_cur, false, bmat, (short)0, c, false, false);

    const float e0 = acc[0] + q2, e1 = acc[1] + q2;
    const float e2 = acc[2] + q2, e3 = acc[3] + q2;
    const float e4 = acc[4] + q2, e5 = acc[5] + q2;
    const float e6 = acc[6] + q2, e7 = acc[7] + q2;

    // Cheap wave-uniform reject: only run insert chains if some lane improves.
    const float m = fminf(fminf(fminf(e0, e1), fminf(e2, e3)),
                          fminf(fminf(e4, e5), fminf(e6, e7)));
    if (__any(m < d2)) {
      const int ib = t * 16 + rbase;
      insert3(e0, ib + 0, d0, d1, d2, i0, i1, i2);
      insert3(e1, ib + 1, d0, d1, d2, i0, i1, i2);
      insert3(e2, ib + 2, d0, d1, d2, i0, i1, i2);
      insert3(e3, ib + 3, d0, d1, d2, i0, i1, i2);
      insert3(e4, ib + 4, d0, d1, d2, i0, i1, i2);
      insert3(e5, ib + 5, d0, d1, d2, i0, i1, i2);
      insert3(e6, ib + 6, d0, d1, d2, i0, i1, i2);
      insert3(e7, ib + 7, d0, d1, d2, i0, i1, i2);
    }
    a_cur = a_nxt;
  }

  // Merge the two lane-halves owning the same query (rows 0-7 vs 8-15).
  float pd0 = __shfl_xor(d0, 16, 32);
  float pd1 = __shfl_xor(d1, 16, 32);
  float pd2 = __shfl_xor(d2, 16, 32);
  int   pi0 = __shfl_xor(i0, 16, 32);
  int   pi1 = __shfl_xor(i1, 16, 32);
  int   pi2 = __shfl_xor(i2, 16, 32);
  insert3(pd0, pi0, d0, d1, d2, i0, i1, i2);
  insert3(pd1, pi1, d0, d1, d2, i0, i1, i2);
  insert3(pd2, pi2, d0, d1, d2, i0, i1, i2);

  if (!hi) {
    // Inverse-distance weights, normalized (dists are squared dists).
    float e0 = fmaxf(d0, 1e-10f);
    float e1 = fmaxf(d1, 1e-10f);
    float e2 = fmaxf(d2, 1e-10f);
    float w0 = 1.0f / (e0 * e0 + 1e-7f);
    float w1 = 1.0f / (e1 * e1 + 1e-7f);
    float w2 = 1.0f / (e2 * e2 + 1e-7f);
    float inv = 1.0f / (w0 + w1 + w2 + 1e-10f);
    w0 *= inv; w1 *= inv; w2 *= inv;
    const size_t o = ((size_t)b * NN + qn) * 3;
    nidx[o + 0] = i0; nidx[o + 1] = i1; nidx[o + 2] = i2;
    nw[o + 0] = w0;   nw[o + 1] = w1;   nw[o + 2] = w2;
  }
}

// Phase 2: weighted gather of sampled_features + concat passthrough features.
// One block per query point: coalesced row reads / row writes.
__global__ __launch_bounds__(CF)
void interp_kernel(const float* __restrict__ feats,    // [B,N,CF]
                   const float* __restrict__ sfeats,   // [B,M,CS]
                   const int* __restrict__ nidx,
                   const float* __restrict__ nw,
                   float* __restrict__ out) {          // [B,N,CS+CF]
  const int q = blockIdx.x;                // b*N + n
  const int b = q / NN;
  const size_t o3 = (size_t)q * 3;
  const int   j0 = nidx[o3 + 0], j1 = nidx[o3 + 1], j2 = nidx[o3 + 2];
  const float w0 = nw[o3 + 0],   w1 = nw[o3 + 1],   w2 = nw[o3 + 2];

  const float* sf = sfeats + (size_t)b * MM * CS;
  const float* f0 = sf + (size_t)j0 * CS;
  const float* f1 = sf + (size_t)j1 * CS;
  const float* f2 = sf + (size_t)j2 * CS;

  const int c = threadIdx.x;               // 0..127
  float a0 = w0 * f0[c]      + w1 * f1[c]      + w2 * f2[c];
  float a1 = w0 * f0[c + CF] + w1 * f1[c + CF] + w2 * f2[c + CF];

  float* op = out + (size_t)q * (CS + CF);
  op[c]      = a0;                          // interp channels 0..127
  op[c + CF] = a1;                          // interp channels 128..255
  op[c + CS] = feats[(size_t)q * CF + c];   // passthrough channels
}

extern "C" void kernel_launch(void* const* d_in, const int* in_sizes, int n_in,
                              void* d_out, int out_size, void* d_ws, size_t ws_size,
                              hipStream_t stream) {
  const float* xyz   = (const float*)d_in[0];   // [B,N,3]
  const float* sxyz  = (const float*)d_in[1];   // [B,M,3]
  const float* feats = (const float*)d_in[2];   // [B,N,128]
  const float* sfeat = (const float*)d_in[3];   // [B,M,256]
  float* out = (float*)d_out;                   // [B,N,384]

  int*   nidx = (int*)d_ws;                                           // 1.5 MB
  float* nw   = (float*)((char*)d_ws + (size_t)BB * NN * 3 * sizeof(int));

  const size_t lds_bytes = (size_t)MM * 3 * 4 + (size_t)MM * 16;  // 48K + 64K
  dim3 g1(BB * (NN / TQ));     // 1024 blocks
  knn_wmma_kernel<<<g1, THREADS, lds_bytes, stream>>>(xyz, sxyz, nidx, nw);

  dim3 g2(BB * NN);            // 131072 blocks, one per query
  interp_kernel<<<g2, CF, 0, stream>>>(feats, sfeat, nidx, nw, out);
}